// LIFFCNeuron_80032420593677
// MI455X (gfx1250) — compile-verified
//
#include <hip/hip_runtime.h>

typedef __attribute__((ext_vector_type(2))) float v2f;
typedef __attribute__((ext_vector_type(4))) float f4;
typedef __attribute__((ext_vector_type(8))) float v8f;

constexpr int NB = 64;      // batch
constexpr int NT = 512;     // timesteps
constexpr int ND = 1024;    // input dim
constexpr int NH = 1024;    // hidden dim
constexpr int HTILE = 64;   // h columns per workgroup
constexpr int KC = 64;      // K chunk streamed through LDS
constexpr int AS = 68;      // LDS row stride (floats) for A chunk (bank-padded, 16B-mult)
constexpr int WSTR = 68;    // LDS row stride for W chunk
constexpr int XS = 65;      // LDS row stride for x buffer (bank-padded)

#define THRESH 0.3f
#define DECAYF (200.0f / 255.0f)

__global__ __launch_bounds__(256) void lif_fc_fused(const float* __restrict__ in,
                                                    const float* __restrict__ Wm,
                                                    const float* __restrict__ bias,
                                                    float* __restrict__ out) {
    extern __shared__ float lds[];
    float* Ab = lds;                 // [NT][AS]   A chunk (input rows)
    float* Wb = lds + NT * AS;       // [HTILE][WSTR] W chunk
    float* Xb = lds;                 // reused after GEMM: [NT][XS] synaptic input

    const int tid  = threadIdx.x;
    const int lane = tid & 31;
    const int wave = tid >> 5;
    const int bb   = blockIdx.x >> 4;            // batch index
    const int h0   = (blockIdx.x & 15) * HTILE;  // h tile base

    const int hsub  = (wave & 3) * 16;           // 16-wide h sub-tile within HTILE
    const int tbase = (wave >> 2) * 256;         // t half owned by this wave
    const int ml    = lane & 15;                 // M / N index within fragment
    const int klo   = (lane >> 4) << 1;          // K offset 0 (lanes 0-15) or 2 (16-31)

    v8f acc[16] = {};                            // 16 t-tiles of 16x16 f32 accumulators

    const float* gA = in + (size_t)bb * NT * ND;

    for (int kc = 0; kc < ND; kc += KC) {
        // --- cooperative load: A chunk 512x64 (float4, fully coalesced) ---
        #pragma unroll
        for (int j = 0; j < 32; ++j) {
            int i = tid + j * 256;               // 0..8191 float4 slots
            int r = i >> 4;                      // t row
            int c = (i & 15) << 2;               // float col
            *(f4*)&Ab[r * AS + c] = *(const f4*)&gA[(size_t)r * ND + kc + c];
        }
        // --- cooperative load: W chunk 64x64 ---
        #pragma unroll
        for (int j = 0; j < 4; ++j) {
            int i = tid + j * 256;               // 0..1023
            int r = i >> 4;
            int c = (i & 15) << 2;
            *(f4*)&Wb[r * WSTR + c] = *(const f4*)&Wm[(size_t)(h0 + r) * ND + kc + c];
        }
        __syncthreads();

        // speculative prefetch of next A chunk (global_prefetch_b8)
        if (kc + KC < ND) {
            int r = tid << 1;                    // rows 0,2,...,510
            __builtin_prefetch(&gA[(size_t)r * ND + kc + KC], 0, 0);
        }

        // --- 16x16x4 f32 WMMA over the chunk; B frag reused across 16 t-tiles ---
        for (int k4 = 0; k4 < KC; k4 += 4) {
            // B fragment: lane n holds B[k][n] = W[h0+hsub+n][kc+k4+k], k = klo..klo+1
            v2f bf = *(const v2f*)&Wb[(hsub + ml) * WSTR + k4 + klo];
            #pragma unroll
            for (int tt = 0; tt < 16; ++tt) {
                // A fragment: lane m holds A[m][k] = in[b][tbase+tt*16+m][kc+k4+k]
                v2f af = *(const v2f*)&Ab[(tbase + tt * 16 + ml) * AS + k4 + klo];
                acc[tt] = __builtin_amdgcn_wmma_f32_16x16x4_f32(
                    false, af, false, bf, (short)0, acc[tt], false, false);
            }
        }
        __syncthreads();
    }

    // --- spill accumulators (D layout: vgpr v -> row v / v+8) into LDS x buffer ---
    #pragma unroll
    for (int tt = 0; tt < 16; ++tt) {
        int trow = tbase + tt * 16 + ((lane >> 4) << 3);
        int n    = hsub + ml;
        #pragma unroll
        for (int v = 0; v < 8; ++v)
            Xb[(trow + v) * XS + n] = acc[tt][v];
    }
    __syncthreads();

    // --- LIF scan: 64 threads, one h-column each, sequential over T ---
    if (tid < HTILE) {
        const int h   = tid;
        const float bv = bias[h0 + h];
        float mem = 0.0f;
        float* op = out + (size_t)bb * NT * NH + h0 + h;
        for (int t = 0; t < NT; ++t) {
            mem += Xb[t * XS + h] + bv;          // integrate
            bool  fire  = mem > THRESH;
            float spike = fire ? 1.0f : 0.0f;    // fire
            mem = fire ? 0.0f : mem * DECAYF;    // reset + decay (BETA = 0)
            op[(size_t)t * NH] = spike;
        }
    }
}

extern "C" void kernel_launch(void* const* d_in, const int* in_sizes, int n_in,
                              void* d_out, int out_size, void* d_ws, size_t ws_size,
                              hipStream_t stream) {
    const float* in   = (const float*)d_in[0];   // [64, 512, 1024]
    const float* Wm   = (const float*)d_in[1];   // [1024, 1024]
    const float* bias = (const float*)d_in[2];   // [1024]
    float* out = (float*)d_out;                  // [64, 512, 1024] spikes

    const size_t shmem = (size_t)(NT * AS + HTILE * WSTR) * sizeof(float); // 156672 B
    hipFuncSetAttribute((const void*)lif_fc_fused,
                        hipFuncAttributeMaxDynamicSharedMemorySize, (int)shmem);

    dim3 grid(NB * (NH / HTILE));                // 1024 workgroups
    dim3 block(256);                             // 8 wave32s
    lif_fc_fused<<<grid, block, shmem, stream>>>(in, Wm, bias, out);
}